// SPS_8048768713426
// MI455X (gfx1250) — compile-verified
//
#include <hip/hip_runtime.h>
#include <math.h>

#define ML_M 256
#define ML_L 256
#define ML_ML (ML_M * ML_L)

__constant__ float kHX = 3.0f;
__constant__ float kHZ = 0.25f;
__constant__ float kJZ = -1.0f;

typedef __attribute__((ext_vector_type(2))) float v2f;
typedef __attribute__((ext_vector_type(8))) float v8f;

// Full-precision f32 wave(32)-wide sum using V_WMMA_F32_16X16X4_F32.
// A(16x4): lane L<16 -> {VGPR0:K=0, VGPR1:K=1} row M=L; lane L>=16 -> {K=2,K=3} row M=L-16.
// With A = {x, 0} and B = all-ones (layout-invariant), D[m,j] = x[m] + x[m+16] for every j.
// C/D(16x16): lane L<16 VGPR r holds (M=r, N=L); lane L>=16 holds (M=8+r, N=L-16).
// Sum of the 8 C VGPRs per lane + xor-16 exchange => every lane holds the full 32-lane sum.
__device__ __forceinline__ float wave_sum_wmma(float x) {
  v2f a; a.x = x;    a.y = 0.0f;
  v2f b; b.x = 1.0f; b.y = 1.0f;
  v8f c = {};
  c = __builtin_amdgcn_wmma_f32_16x16x4_f32(false, a, false, b, (short)0, c,
                                            false, false);
  float r = ((c[0] + c[1]) + (c[2] + c[3])) + ((c[4] + c[5]) + (c[6] + c[7]));
  r += __shfl_xor(r, 16, 32);
  return r;
}

__device__ __forceinline__ float wave_prod(float x) {
#pragma unroll
  for (int m = 1; m < 32; m <<= 1) x *= __shfl_xor(x, m, 32);
  return x;
}

// ---------------------------------------------------------------- prep: sincos
__global__ __launch_bounds__(256) void vmc_prep(const float* __restrict__ theta,
                                                float* __restrict__ cs, int n) {
  int i = blockIdx.x * blockDim.x + threadIdx.x;
  if (i < n) {
    float s, c;
    sincosf(theta[i], &s, &c);
    cs[2 * i]     = c;
    cs[2 * i + 1] = s;
  }
}

// ---------------------------------------------------------------- main fused pass
// grid = (M, C). Block handles row m, b in [c*bcount, (c+1)*bcount). thread = l.
__global__ __launch_bounds__(256) void vmc_main(
    const float* __restrict__ cs,         // (M*L) float2 interleaved cos,sin
    const float* __restrict__ coef,       // (M)
    const float* __restrict__ strengths,  // (L-1)
    float* __restrict__ pN,               // (C, M*L) partial 2*sum_b base*(...)
    float* __restrict__ pD,               // (C, M*L) partial 2*sum_b base*g_d
    float* __restrict__ pa1,              // (C, M) partial sum_b coef[b]*P*F
    float* __restrict__ pa2,              // (C, M) partial sum_b coef[b]*P
    int bcount)
{
  __shared__ float a_sh[ML_L + 2];        // padded: a_sh[0] = a_sh[L+1] = 0
  __shared__ float red_sh[8][4];

  const int l    = threadIdx.x;
  const int m    = blockIdx.x;
  const int c    = blockIdx.y;
  const int wave = l >> 5;
  const int lane = l & 31;
  const int blo  = c * bcount;
  const int bhi  = blo + bcount;

  if (l == 0) a_sh[0] = 0.0f;
  if (l == 1) a_sh[ML_L + 1] = 0.0f;

  const float2 csm = reinterpret_cast<const float2*>(cs)[m * ML_L + l];
  const float cm = csm.x, sm = csm.y;
  const float s_own  = (l < ML_L - 1) ? strengths[l]     : 0.0f;
  const float s_prev = (l > 0)        ? strengths[l - 1] : 0.0f;
  const float coefm = coef[m];
  const float hx = kHX, hz = kHZ, jz = kJZ;

  float accN = 0.0f, accD = 0.0f, a1 = 0.0f, a2 = 0.0f;

  for (int b = blo; b < bhi; ++b) {
    const float2 csb = reinterpret_cast<const float2*>(cs)[b * ML_L + l];
    const float cb = csb.x, sb = csb.y;

    const float cosd = cm * cb + sm * sb;   // cos(tm - tb)
    const float sind = sm * cb - cm * sb;   // sin(tm - tb)
    const float sins = sm * cb + cm * sb;   // sin(tm + tb)
    const float coss = cm * cb - sm * sb;   // cos(tm + tb)
    const float inv  = 1.0f / (cosd + 1e-12f);
    const float av   = coss * inv;          // a

    a_sh[l + 1] = av;
    __syncthreads();
    const float a_next = a_sh[l + 2];       // 0 at l = L-1 (padding)
    const float a_prev = a_sh[l];           // 0 at l = 0   (padding)

    // per-l partials for the L-reductions
    const float pU = sins * inv;
    const float pV = av;                    // coss*inv
    const float pS = s_own * av * a_next;   // edge (l, l+1)

    // wave-level sums on the matrix unit, product on shuffles
    const float wU = wave_sum_wmma(pU);
    const float wV = wave_sum_wmma(pV);
    const float wS = wave_sum_wmma(pS);
    const float wP = wave_prod(cosd);

    if (lane == 0) {
      red_sh[wave][0] = wU;
      red_sh[wave][1] = wV;
      red_sh[wave][2] = wS;
      red_sh[wave][3] = wP;
    }
    __syncthreads();
    float U = 0.0f, V = 0.0f, S = 0.0f, P = 1.0f;
#pragma unroll
    for (int w = 0; w < 8; ++w) {
      U += red_sh[w][0];
      V += red_sh[w][1];
      S += red_sh[w][2];
      P *= red_sh[w][3];
    }

    const float F     = hz * V + hx * U + jz * S;
    const float coefb = coef[b];
    const float base  = coefm * coefb * P;
    a1 += coefb * P * F;   // -> dN_dc / 2 (per m)
    a2 += coefb * P;       // -> dD_dc / 2 (per m)

    // backward per-(b,l) terms
    const float g_d   = -sind * inv;
    const float sinv2 = sind * inv * inv;
    const float Gxz   = (-hz * sins + hx * coss) * inv +
                        (hz * coss + hx * sins) * sinv2;
    const float inc   = s_own * a_next + s_prev * a_prev;
    const float Gzz   = (-sins * inv + coss * sinv2) * inc * jz;

    accD += base * g_d;
    accN += base * (g_d * F + Gxz + Gzz);
  }

  pN[c * ML_ML + m * ML_L + l] = 2.0f * accN;
  pD[c * ML_ML + m * ML_L + l] = 2.0f * accD;
  if (l == 0) {
    pa1[c * ML_M + m] = a1;
    pa2[c * ML_M + m] = a2;
  }
}

// ---------------------------------------------------------------- combine chunks (theta)
__global__ __launch_bounds__(256) void vmc_combine(const float* __restrict__ pN,
                                                   const float* __restrict__ pD,
                                                   float* __restrict__ dN_th,
                                                   float* __restrict__ dD_th,
                                                   int C, int n) {
  const int i = blockIdx.x * blockDim.x + threadIdx.x;
  if (i < n) {
    float sn = 0.0f, sd = 0.0f;
    for (int c = 0; c < C; ++c) {
      sn += pN[c * n + i];
      sd += pD[c * n + i];
    }
    dN_th[i] = sn;
    dD_th[i] = sd;
  }
}

// ---------------------------------------------------------------- scalars: N, D, E, dE_dc
__global__ __launch_bounds__(256) void vmc_scalars(
    const float* __restrict__ coef,
    const float* __restrict__ pa1,   // (C, M)
    const float* __restrict__ pa2,   // (C, M)
    int C,
    float* __restrict__ dE_dc,   // d_out + ML  (M)
    float* __restrict__ E_out,   // d_out + ML + M (1)
    float* __restrict__ ED_ws)   // ws: [0]=E, [1]=D
{
  __shared__ float sN[ML_M];
  __shared__ float sD[ML_M];
  __shared__ float Esh, Dsh;
  const int t = threadIdx.x;

  float a1 = 0.0f, a2 = 0.0f;
  for (int c = 0; c < C; ++c) {
    a1 += pa1[c * ML_M + t];
    a2 += pa2[c * ML_M + t];
  }

  const float cm = coef[t];
  sN[t] = cm * a1;
  sD[t] = cm * a2;
  __syncthreads();
#pragma unroll
  for (int s = ML_M / 2; s > 0; s >>= 1) {
    if (t < s) {
      sN[t] += sN[t + s];
      sD[t] += sD[t + s];
    }
    __syncthreads();
  }
  if (t == 0) {
    const float N = sN[0], D = sD[0];
    const float E = N / D;
    Esh = E;
    Dsh = D;
    ED_ws[0] = E;
    ED_ws[1] = D;
    E_out[0] = E / (float)ML_L;
  }
  __syncthreads();
  const float E = Esh, D = Dsh;
  dE_dc[t] = (2.0f * a1 - E * 2.0f * a2) / D;
}

// ---------------------------------------------------------------- theta combine (in place)
__global__ __launch_bounds__(256) void vmc_theta(float* __restrict__ dN_th,
                                                 const float* __restrict__ dD_th,
                                                 const float* __restrict__ ED,
                                                 int n) {
  const int i = blockIdx.x * blockDim.x + threadIdx.x;
  if (i < n) {
    const float E = ED[0], D = ED[1];
    dN_th[i] = (dN_th[i] - E * dD_th[i]) / D;
  }
}

extern "C" void kernel_launch(void* const* d_in, const int* in_sizes, int n_in,
                              void* d_out, int out_size, void* d_ws, size_t ws_size,
                              hipStream_t stream) {
  const float* theta     = (const float*)d_in[0];  // (M,L)
  const float* coef      = (const float*)d_in[1];  // (M)
  // d_in[2]/d_in[3] are the chain edges (u=l, v=l+1 by construction) — folded in.
  const float* strengths = (const float*)d_in[4];  // (L-1)

  float* out = (float*)d_out;
  float* ws  = (float*)d_ws;

  // Base ws layout (floats):
  float* dD_th = ws;                       // ML
  float* cs    = ws + ML_ML;               // 2*ML  (cos,sin interleaved)
  float* ED    = ws + 3 * ML_ML;           // 2
  float* base_pa1 = ED + 2;                // M  (single-chunk fallback)
  float* base_pa2 = base_pa1 + ML_M;       // M
  const size_t base_floats = (size_t)3 * ML_ML + 2 + 2 * ML_M;

  // Pick chunk count C from available scratch (deterministic for a given run):
  // extra need = C * (2*ML + 2*M) floats.
  int C = 1;
  for (int cand = 8; cand >= 2; cand >>= 1) {
    size_t need = (base_floats + (size_t)cand * (2 * ML_ML + 2 * ML_M)) * sizeof(float);
    if (ws_size >= need) { C = cand; break; }
  }

  float* dN_th = out;                      // first ML of output, combined in place
  float* dE_dc = out + ML_ML;              // M
  float* E_out = out + ML_ML + ML_M;       // 1

  float *pN, *pD, *pa1, *pa2;
  if (C > 1) {
    float* ext = ws + base_floats;
    pN  = ext;                             // C*ML
    pD  = ext + (size_t)C * ML_ML;         // C*ML
    pa1 = pD + (size_t)C * ML_ML;          // C*M
    pa2 = pa1 + (size_t)C * ML_M;          // C*M
  } else {
    pN  = dN_th;                           // write finals directly
    pD  = dD_th;
    pa1 = base_pa1;
    pa2 = base_pa2;
  }

  vmc_prep<<<ML_ML / 256, 256, 0, stream>>>(theta, cs, ML_ML);
  vmc_main<<<dim3(ML_M, C), ML_L, 0, stream>>>(cs, coef, strengths,
                                               pN, pD, pa1, pa2, ML_M / C);
  if (C > 1) {
    vmc_combine<<<ML_ML / 256, 256, 0, stream>>>(pN, pD, dN_th, dD_th, C, ML_ML);
  }
  vmc_scalars<<<1, ML_M, 0, stream>>>(coef, pa1, pa2, C, dE_dc, E_out, ED);
  vmc_theta<<<ML_ML / 256, 256, 0, stream>>>(dN_th, dD_th, ED, ML_ML);
}